// TextRGCN_56341380989101
// MI455X (gfx1250) — compile-verified
//
#include <hip/hip_runtime.h>
#include <hip/hip_bf16.h>

// ---------------------------------------------------------------------------
// TextRGCN pipeline for gfx1250 (MI455X), fp32 WMMA (V_WMMA_F32_16X16X4_F32).
// ---------------------------------------------------------------------------

typedef float v2f __attribute__((ext_vector_type(2)));
typedef float v4f __attribute__((ext_vector_type(4)));
typedef float v8f __attribute__((ext_vector_type(8)));

#define H 256
#define TD 768
#define NB 8
#define XB_LD (9 * H)        // 8 bases + root slice
#define WAVES_PER_BLOCK 8
#define ROWS_PER_BLOCK (WAVES_PER_BLOCK * 16)   // 128
#define LN_EPS 1e-5f

__device__ __forceinline__ v8f wmma_f32_k4(v2f a, v2f b, v8f c) {
    // D(16x16,f32) = A(16x4,f32) x B(4x16,f32) + C
    return __builtin_amdgcn_wmma_f32_16x16x4_f32(
        /*neg_a=*/false, a, /*neg_b=*/false, b,
        /*c_mod=*/(short)0, c, /*reuse_a=*/false, /*reuse_b=*/false);
}

// ---------------------------------------------------------------------------
// generic zero fill (vectorized, n must be multiple of 4)
__global__ void zero_kernel(v4f* __restrict__ p, long n4) {
    long i = (long)blockIdx.x * blockDim.x + threadIdx.x;
    long stride = (long)gridDim.x * blockDim.x;
    v4f z = {};
    for (; i < n4; i += stride) p[i] = z;
}

// ---------------------------------------------------------------------------
// degree histogram: deg[dst[e]] += 1
__global__ void deg_kernel(const int* __restrict__ dst, float* __restrict__ deg, int E) {
    int e = blockIdx.x * blockDim.x + threadIdx.x;
    if (e < E) unsafeAtomicAdd(&deg[dst[e]], 1.0f);
}

// ---------------------------------------------------------------------------
// x0[n,h] = ent[n,h] + sum_d tb[n,d] * w[h,d] + pb[h]
// A = tb [M,768]; B(k,n) = w[n*768+k]  (per-lane contiguous float2 loads)
__global__ void proj_gemm_kernel(const float* __restrict__ tb,
                                 const float* __restrict__ w,
                                 const float* __restrict__ pb,
                                 const float* __restrict__ ent,
                                 float* __restrict__ out,
                                 int M) {
    const int lane = threadIdx.x & 31;
    const int wv   = threadIdx.x >> 5;
    const int l15  = lane & 15;
    const int lhi  = lane >> 4;                 // 0 or 1

    const int rowBase = (blockIdx.x * WAVES_PER_BLOCK + wv) * 16;
    if (rowBase >= M) return;                   // wave-uniform exit (EXEC all ones)
    const int colBase = blockIdx.y * 64;        // gridDim.y = 256/64 = 4

    v8f acc[4] = {};
    const float* aRow = tb + (size_t)(rowBase + l15) * TD + 2 * lhi;
    const float* bCol = w  + (size_t)(colBase + l15) * TD + 2 * lhi;

    for (int k = 0; k < TD; k += 4) {
        v2f a = *(const v2f*)(aRow + k);
#pragma unroll
        for (int j = 0; j < 4; ++j) {
            v2f b = *(const v2f*)(bCol + (size_t)j * 16 * TD + k);
            acc[j] = wmma_f32_k4(a, b, acc[j]);
        }
    }
#pragma unroll
    for (int j = 0; j < 4; ++j) {
        int col = colBase + j * 16 + l15;
#pragma unroll
        for (int r = 0; r < 8; ++r) {
            int row = rowBase + r + 8 * lhi;
            size_t idx = (size_t)row * H + col;
            out[idx] = acc[j][r] + ent[idx] + pb[col];
        }
    }
}

// ---------------------------------------------------------------------------
// in-place LayerNorm over H=256, one wave per row
__global__ void ln_kernel(float* __restrict__ x,
                          const float* __restrict__ g,
                          const float* __restrict__ b, int M) {
    const int lane = threadIdx.x & 31;
    const int wv   = threadIdx.x >> 5;
    const int row  = blockIdx.x * WAVES_PER_BLOCK + wv;
    if (row >= M) return;
    float* xr = x + (size_t)row * H;

    float v[8];
    float s = 0.0f;
#pragma unroll
    for (int i = 0; i < 8; ++i) { v[i] = xr[lane + 32 * i]; s += v[i]; }
#pragma unroll
    for (int m = 16; m > 0; m >>= 1) s += __shfl_xor(s, m, 32);
    float mu = s * (1.0f / H);

    float q = 0.0f;
#pragma unroll
    for (int i = 0; i < 8; ++i) { float d = v[i] - mu; q += d * d; }
#pragma unroll
    for (int m = 16; m > 0; m >>= 1) q += __shfl_xor(q, m, 32);
    float inv = rsqrtf(q * (1.0f / H) + LN_EPS);

#pragma unroll
    for (int i = 0; i < 8; ++i) {
        int c = lane + 32 * i;
        xr[c] = g[c] * (v[i] - mu) * inv + b[c];
    }
}

// ---------------------------------------------------------------------------
// xb[n, s*256 + f] = sum_d x[n,d] * B_s[d,f], s=0..7 -> basis[s], s=8 -> root
// B stored [K=256, N=256] row-major -> two strided scalar loads per lane
// (lanes 0-15 cover 16 consecutive floats per load -> full 64B segments, L2-hot)
__global__ void basis_gemm_kernel(const float* __restrict__ x,
                                  const float* __restrict__ basis,
                                  const float* __restrict__ root,
                                  float* __restrict__ xb,
                                  int M) {
    const int lane = threadIdx.x & 31;
    const int wv   = threadIdx.x >> 5;
    const int l15  = lane & 15;
    const int lhi  = lane >> 4;

    const int rowBase = (blockIdx.x * WAVES_PER_BLOCK + wv) * 16;
    if (rowBase >= M) return;

    const int g        = blockIdx.y;        // 0..35, each covers 64 output cols
    const int slice    = g >> 2;            // 0..8 (4 blocks of 64 cols per slice)
    const int nInSlice = (g & 3) * 64;
    const float* B = (slice < NB) ? (basis + (size_t)slice * H * H) : root;

    v8f acc[4] = {};
    const float* aRow  = x + (size_t)(rowBase + l15) * H + 2 * lhi;
    const float* bBase = B + nInSlice + l15;

    for (int k = 0; k < H; k += 4) {
        v2f a = *(const v2f*)(aRow + k);
        int kk = k + 2 * lhi;
#pragma unroll
        for (int j = 0; j < 4; ++j) {
            v2f b;
            b.x = bBase[(size_t)kk * H + j * 16];
            b.y = bBase[(size_t)(kk + 1) * H + j * 16];
            acc[j] = wmma_f32_k4(a, b, acc[j]);
        }
    }
#pragma unroll
    for (int j = 0; j < 4; ++j) {
        int col = g * 64 + j * 16 + l15;
#pragma unroll
        for (int r = 0; r < 8; ++r) {
            int row = rowBase + r + 8 * lhi;
            xb[(size_t)row * XB_LD + col] = acc[j][r];
        }
    }
}

// ---------------------------------------------------------------------------
// per-edge message + scatter: agg[dst,f] += sum_b comp[etype,b] * xb[src,b,f]
// 64 threads per edge, float4 (b128) gathers; 4 edges per 256-thread block.
__global__ void edge_kernel(const int* __restrict__ src,
                            const int* __restrict__ dst,
                            const int* __restrict__ etype,
                            const float* __restrict__ comp,   // [R2, NB]
                            const float* __restrict__ xb,     // [N, 9*256]
                            float* __restrict__ agg,          // [N, 256]
                            int E) {
    const int grp    = threadIdx.x >> 6;          // 0..3
    const int lane64 = threadIdx.x & 63;
    const int e      = blockIdx.x * 4 + grp;
    const bool valid = e < E;

    __shared__ float cs[4][NB];
    if (valid && lane64 < NB) cs[grp][lane64] = comp[(size_t)etype[e] * NB + lane64];
    __syncthreads();
    if (!valid) return;

    const int f0 = lane64 * 4;                    // 0..252, 16B aligned
    const float* xp = xb + (size_t)src[e] * XB_LD + f0;

    v4f acc = {};
#pragma unroll
    for (int b = 0; b < NB; ++b) {
        v4f v = *(const v4f*)(xp + b * H);
        float c = cs[grp][b];
        acc.x += c * v.x; acc.y += c * v.y; acc.z += c * v.z; acc.w += c * v.w;
    }
    float* ap = agg + (size_t)dst[e] * H + f0;
    unsafeAtomicAdd(ap + 0, acc.x);
    unsafeAtomicAdd(ap + 1, acc.y);
    unsafeAtomicAdd(ap + 2, acc.z);
    unsafeAtomicAdd(ap + 3, acc.w);
}

// ---------------------------------------------------------------------------
// x_next = relu(agg * inv_deg + xb[:,8,:] + bias)   (float4 vectorized)
__global__ void combine_kernel(const float* __restrict__ agg,
                               const float* __restrict__ xb,
                               const float* __restrict__ deg,
                               const float* __restrict__ bias,
                               float* __restrict__ xout,
                               int M) {
    long idx4 = (long)blockIdx.x * blockDim.x + threadIdx.x;   // over M*H/4
    if (idx4 >= (long)M * (H / 4)) return;
    int n  = (int)(idx4 >> 6);            // H/4 = 64 vectors per row
    int f0 = (int)(idx4 & 63) * 4;
    float inv = 1.0f / fmaxf(deg[n], 1.0f);

    v4f a  = *(const v4f*)(agg + (size_t)n * H + f0);
    v4f xr = *(const v4f*)(xb + (size_t)n * XB_LD + NB * H + f0);
    v4f bb = *(const v4f*)(bias + f0);
    v4f r;
    r.x = fmaxf(a.x * inv + xr.x + bb.x, 0.0f);
    r.y = fmaxf(a.y * inv + xr.y + bb.y, 0.0f);
    r.z = fmaxf(a.z * inv + xr.z + bb.z, 0.0f);
    r.w = fmaxf(a.w * inv + xr.w + bb.w, 0.0f);
    *(v4f*)(xout + (size_t)n * H + f0) = r;
}

// ---------------------------------------------------------------------------
// ComplEx-style scoring, one wave per triple
__global__ void score_kernel(const float* __restrict__ x,       // [N,256]
                             const float* __restrict__ rel,     // [R,256]
                             const int* __restrict__ heads,
                             const int* __restrict__ rels,
                             const int* __restrict__ tails,
                             float* __restrict__ out, int T) {
    const int lane = threadIdx.x & 31;
    const int wv   = threadIdx.x >> 5;
    const int t    = blockIdx.x * WAVES_PER_BLOCK + wv;
    if (t >= T) return;

    const float* h  = x   + (size_t)heads[t] * H;
    const float* tl = x   + (size_t)tails[t] * H;
    const float* r  = rel + (size_t)rels[t]  * H;

    float s = 0.0f;
#pragma unroll
    for (int i0 = 0; i0 < 4; ++i0) {
        int i = lane + 32 * i0;                     // 0..127
        float hr = h[i],  hi = h[i + 128];
        float tr = tl[i], ti = tl[i + 128];
        float rr = r[i],  ri = r[i + 128];
        s += hr * rr * tr + hi * rr * ti + hr * ri * ti - hi * ri * tr;
    }
#pragma unroll
    for (int m = 16; m > 0; m >>= 1) s += __shfl_xor(s, m, 32);
    if (lane == 0) out[t] = s;
}

// ---------------------------------------------------------------------------
extern "C" void kernel_launch(void* const* d_in, const int* in_sizes, int n_in,
                              void* d_out, int out_size, void* d_ws, size_t ws_size,
                              hipStream_t stream) {
    const float* ent_emb   = (const float*)d_in[0];
    const float* text_base = (const float*)d_in[1];
    const float* text_w    = (const float*)d_in[2];
    const float* text_b    = (const float*)d_in[3];
    const float* ln_g      = (const float*)d_in[4];
    const float* ln_b      = (const float*)d_in[5];
    const float* comp1     = (const float*)d_in[6];
    const float* basis1    = (const float*)d_in[7];
    const float* root1     = (const float*)d_in[8];
    const float* bias1     = (const float*)d_in[9];
    const float* comp2     = (const float*)d_in[10];
    const float* basis2    = (const float*)d_in[11];
    const float* root2     = (const float*)d_in[12];
    const float* bias2     = (const float*)d_in[13];
    const float* rel_emb   = (const float*)d_in[14];
    const int* edge_src    = (const int*)d_in[15];
    const int* edge_dst    = (const int*)d_in[16];
    const int* edge_type   = (const int*)d_in[17];
    const int* heads       = (const int*)d_in[18];
    const int* rels        = (const int*)d_in[19];
    const int* tails       = (const int*)d_in[20];

    const int N = in_sizes[0] / H;     // 50000
    const int E = in_sizes[15];        // 300000
    const int T = in_sizes[18];        // 8192

    // workspace carve-out
    char* ws = (char*)d_ws;
    size_t off = 0;
    auto carve = [&](size_t bytes) {
        char* p = ws + off;
        off = (off + bytes + 255) & ~(size_t)255;
        return (float*)p;
    };
    float* XB  = carve((size_t)N * XB_LD * sizeof(float));  // 460.8 MB
    float* X0  = carve((size_t)N * H * sizeof(float));      // 51.2 MB
    float* X1  = carve((size_t)N * H * sizeof(float));      // 51.2 MB
    float* AGG = carve((size_t)N * H * sizeof(float));      // 51.2 MB
    float* DEG = carve((size_t)N * sizeof(float));          // 0.2 MB
    (void)ws_size;

    const dim3 blk(256);
    const int rowBlocks = (N + ROWS_PER_BLOCK - 1) / ROWS_PER_BLOCK;   // 391

    // degrees (shared by both layers)
    zero_kernel<<<dim3(64), blk, 0, stream>>>((v4f*)DEG, (long)N / 4);
    deg_kernel<<<dim3((E + 255) / 256), blk, 0, stream>>>(edge_dst, DEG, E);

    // x = LN(ent + tb @ W^T + b)
    proj_gemm_kernel<<<dim3(rowBlocks, H / 64), blk, 0, stream>>>(
        text_base, text_w, text_b, ent_emb, X0, N);
    ln_kernel<<<dim3((N + WAVES_PER_BLOCK - 1) / WAVES_PER_BLOCK), blk, 0, stream>>>(
        X0, ln_g, ln_b, N);

    // ---- layer 1: X0 -> X1 ----
    basis_gemm_kernel<<<dim3(rowBlocks, XB_LD / 64), blk, 0, stream>>>(
        X0, basis1, root1, XB, N);
    zero_kernel<<<dim3(2048), blk, 0, stream>>>((v4f*)AGG, (long)N * H / 4);
    edge_kernel<<<dim3((E + 3) / 4), blk, 0, stream>>>(
        edge_src, edge_dst, edge_type, comp1, XB, AGG, E);
    combine_kernel<<<dim3((N * (H / 4) + 255) / 256), blk, 0, stream>>>(
        AGG, XB, DEG, bias1, X1, N);

    // ---- layer 2: X1 -> X0 (buffer reuse) ----
    basis_gemm_kernel<<<dim3(rowBlocks, XB_LD / 64), blk, 0, stream>>>(
        X1, basis2, root2, XB, N);
    zero_kernel<<<dim3(2048), blk, 0, stream>>>((v4f*)AGG, (long)N * H / 4);
    edge_kernel<<<dim3((E + 3) / 4), blk, 0, stream>>>(
        edge_src, edge_dst, edge_type, comp2, XB, AGG, E);
    combine_kernel<<<dim3((N * (H / 4) + 255) / 256), blk, 0, stream>>>(
        AGG, XB, DEG, bias2, X0, N);

    // ---- scoring ----
    score_kernel<<<dim3((T + WAVES_PER_BLOCK - 1) / WAVES_PER_BLOCK), blk, 0, stream>>>(
        X0, rel_emb, heads, rels, tails, (float*)d_out, T);
}